// GCEncoder_42125039239628
// MI455X (gfx1250) — compile-verified
//
#include <hip/hip_runtime.h>

typedef __attribute__((ext_vector_type(2))) float v2f;
typedef __attribute__((ext_vector_type(8))) float v8f;

#define TILE_M 64
#define TILE_N 128
#define TILE_K 32
#define LDA_S  36    // A tile LDS row stride (floats): even -> b64 frags, conflict-free
#define LDB_KP 288   // B tile LDS k-pair stride (floats): 288 % 64 == 32 -> disjoint halves

// ---------------------------------------------------------------------------
// f32 WMMA GEMM: C[M,N] = op(A)[M,K] * B[K,N]   (op = identity or ReLU)
// 256 threads = 8 waves (4 M x 2 N); wave tile 16x64 (4 accumulators);
// WG tile 64x128; K staged through LDS in steps of 32.
// A LDS layout: sA[r*36 + k]                (fragment = ds_load_b64)
// B LDS layout: sB[(k>>1)*288 + 2*col + (k&1)] (fragment = ds_load_b64)
// Requires K%32==0, N%128==0, M%16==0 (store granularity).
// ---------------------------------------------------------------------------
template <bool RELU>
__global__ __launch_bounds__(256) void gemm_f32_wmma(
    const float* __restrict__ A, const float* __restrict__ B,
    float* __restrict__ C, int M, int N, int K)
{
    __shared__ float sA[TILE_M * LDA_S];          //  9216 B
    __shared__ float sB[(TILE_K / 2) * LDB_KP];   // 18432 B

    const int tid   = threadIdx.x;
    const int lane  = tid & 31;
    const int wid   = tid >> 5;
    const int waveM = wid >> 1;   // 0..3
    const int waveN = wid & 1;    // 0..1

    const int blockM = blockIdx.x * TILE_M;
    const int blockN = blockIdx.y * TILE_N;
    const int m0 = blockM + waveM * 16;
    const int n0 = blockN + waveN * 64;

    v8f acc0 = {}, acc1 = {}, acc2 = {}, acc3 = {};

    // A fragment (16x4 f32): lanes 0-15 hold K+0/K+1, lanes 16-31 hold K+2/K+3
    const int aRow = waveM * 16 + (lane & 15);
    const int kHi  = (lane >> 4) * 2;                                  // 0 or 2
    const float* aP = &sA[aRow * LDA_S];
    // B fragment: lane half selects k-pair (p or p+1); column = n-base + lane&15
    const float* bP = &sB[(lane >> 4) * LDB_KP + 2 * (waveN * 64 + (lane & 15))];

    for (int k0 = 0; k0 < K; k0 += TILE_K) {
        __syncthreads();
        // ---- stage A tile (64 x 32) as float4, row-clamped at M tail
        #pragma unroll
        for (int i = 0; i < 2; ++i) {
            int idx = tid + i * 256;              // 0..511
            int r = idx >> 3, c = (idx & 7) * 4;
            int gr = blockM + r;
            if (gr >= M) gr = M - 1;
            float4 v = *(const float4*)&A[(size_t)gr * K + (k0 + c)];
            if (RELU) {
                v.x = fmaxf(v.x, 0.0f); v.y = fmaxf(v.y, 0.0f);
                v.z = fmaxf(v.z, 0.0f); v.w = fmaxf(v.w, 0.0f);
            }
            *(float4*)&sA[r * LDA_S + c] = v;
        }
        // ---- stage B tile (32 x 128) k-pair interleaved, two rows per item
        #pragma unroll
        for (int i = 0; i < 2; ++i) {
            int idx = tid + i * 256;              // 0..511
            int kp = idx >> 5, col = (idx & 31) * 4;
            const float* g0 = &B[(size_t)(k0 + 2 * kp) * N + (blockN + col)];
            float4 r0 = *(const float4*)g0;
            float4 r1 = *(const float4*)(g0 + N);
            float* d = &sB[kp * LDB_KP + 2 * col];
            *(float4*)(d)     = make_float4(r0.x, r1.x, r0.y, r1.y);
            *(float4*)(d + 4) = make_float4(r0.z, r1.z, r0.w, r1.w);
        }
        __syncthreads();
        // ---- compute: 8 k-steps x 4 n-fragments = 32 WMMAs per tile
        #pragma unroll
        for (int kk = 0; kk < TILE_K; kk += 4) {
            v2f a = *(const v2f*)(aP + kk + kHi);
            const float* bk = bP + (kk >> 1) * LDB_KP;
            v2f b0 = *(const v2f*)(bk);
            acc0 = __builtin_amdgcn_wmma_f32_16x16x4_f32(false, a, false, b0, (short)0, acc0, false, false);
            v2f b1 = *(const v2f*)(bk + 32);
            acc1 = __builtin_amdgcn_wmma_f32_16x16x4_f32(false, a, false, b1, (short)0, acc1, false, false);
            v2f b2 = *(const v2f*)(bk + 64);
            acc2 = __builtin_amdgcn_wmma_f32_16x16x4_f32(false, a, false, b2, (short)0, acc2, false, false);
            v2f b3 = *(const v2f*)(bk + 96);
            acc3 = __builtin_amdgcn_wmma_f32_16x16x4_f32(false, a, false, b3, (short)0, acc3, false, false);
        }
    }

    // C/D layout: VGPR i -> row = i + 8*(lane>>4), col = lane&15
    if (m0 < M) {                       // wave-uniform guard (M % 16 == 0)
        const int rbase = m0 + (lane >> 4) * 8;
        const int cidx  = lane & 15;
        #pragma unroll
        for (int i = 0; i < 8; ++i) {
            size_t rowOff = (size_t)(rbase + i) * N;
            C[rowOff + n0 +  0 + cidx] = acc0[i];
            C[rowOff + n0 + 16 + cidx] = acc1[i];
            C[rowOff + n0 + 32 + cidx] = acc2[i];
            C[rowOff + n0 + 48 + cidx] = acc3[i];
        }
    }
}

// Hardware f32 atomic add (non-returning): guaranteed global_atomic_add_f32.
__device__ __forceinline__ void atomic_fadd_f32(float* p, float v)
{
    asm volatile("global_atomic_add_f32 %0, %1, off"
                 :: "v"(p), "v"(v)
                 : "memory");
}

// ---------------------------------------------------------------------------
// Edge scatter: agg[dst[e]] += w[e] * s[src[e]]   (float4 per thread)
// d4shift = log2(D/4) >= 5, so e = gid>>d4shift is wave-uniform -> scalarize.
// ---------------------------------------------------------------------------
__global__ __launch_bounds__(256) void scatter_edges(
    const float* __restrict__ s, const int* __restrict__ src,
    const int* __restrict__ dst, const float* __restrict__ w,
    float* __restrict__ agg, int nE, int d4shift)
{
    int gid = blockIdx.x * blockDim.x + threadIdx.x;
    int total = nE << d4shift;
    if (gid >= total) return;
    int e = __builtin_amdgcn_readfirstlane(gid >> d4shift);  // wave-uniform edge id
    int c = gid & ((1 << d4shift) - 1);
    int sn = src[e];
    int dn = dst[e];
    float wt = w[e];
    const float4* sp = (const float4*)s + ((size_t)sn << d4shift);
    float4 v = sp[c];
    float* op = agg + (((size_t)dn << d4shift) + (size_t)c) * 4;
    atomic_fadd_f32(op + 0, v.x * wt);
    atomic_fadd_f32(op + 1, v.y * wt);
    atomic_fadd_f32(op + 2, v.z * wt);
    atomic_fadd_f32(op + 3, v.w * wt);
}

// out[i] = bias[i & dmask]  (D is a power of two)
__global__ __launch_bounds__(256) void init_bias(
    float* __restrict__ out, const float* __restrict__ bias, int total, int dmask)
{
    int gid = blockIdx.x * blockDim.x + threadIdx.x;
    if (gid < total) out[gid] = bias[gid & dmask];
}

__global__ __launch_bounds__(256) void relu_inplace(float* __restrict__ p, int total)
{
    int gid = blockIdx.x * blockDim.x + threadIdx.x;
    if (gid < total) p[gid] = fmaxf(p[gid], 0.0f);
}

// ---------------------------------------------------------------------------
extern "C" void kernel_launch(void* const* d_in, const int* in_sizes, int n_in,
                              void* d_out, int out_size, void* d_ws, size_t ws_size,
                              hipStream_t stream)
{
    const float* x  = (const float*)d_in[0];
    const int*   ei = (const int*)d_in[1];
    const float* ew = (const float*)d_in[2];
    const float* W1 = (const float*)d_in[3];
    const float* b1 = (const float*)d_in[4];
    const float* W2 = (const float*)d_in[5];
    const float* b2 = (const float*)d_in[6];
    float* out = (float*)d_out;

    const int D_HID = in_sizes[4];           // 512
    const int D_LAT = in_sizes[6];           // 256
    const int D_IN  = in_sizes[3] / D_HID;   // 1024
    const int Nn    = in_sizes[0] / D_IN;    // 50000 nodes
    const int nE    = in_sizes[2];           // 400000 edges
    const int* src = ei;
    const int* dst = ei + nE;

    // workspace: s_buf (Nn*D_HID f32, reused for s2) | agg1 (Nn*D_HID f32)
    float* s_buf = (float*)d_ws;
    float* agg1  = s_buf + (size_t)Nn * D_HID;

    dim3 blk(256);

    // 1) agg1 = b1 broadcast (bias folded into aggregation init)
    {
        int total = Nn * D_HID;
        init_bias<<<dim3((total + 255) / 256), blk, 0, stream>>>(agg1, b1, total, D_HID - 1);
    }
    // 2) s1 = x @ W1
    {
        dim3 grid((Nn + TILE_M - 1) / TILE_M, D_HID / TILE_N);
        gemm_f32_wmma<false><<<grid, blk, 0, stream>>>(x, W1, s_buf, Nn, D_HID, D_IN);
    }
    // 3) agg1[dst] += w * s1[src]
    {
        int d4shift = __builtin_ctz(D_HID >> 2);           // 7
        int total = nE << d4shift;
        scatter_edges<<<dim3((total + 255) / 256), blk, 0, stream>>>(
            s_buf, src, dst, ew, agg1, nE, d4shift);
    }
    // 4) s2 = relu(agg1) @ W2  (ReLU fused into A-operand load; reuses s_buf)
    {
        dim3 grid((Nn + TILE_M - 1) / TILE_M, D_LAT / TILE_N);
        gemm_f32_wmma<true><<<grid, blk, 0, stream>>>(agg1, W2, s_buf, Nn, D_LAT, D_HID);
    }
    // 5) out = b2 broadcast
    {
        int total = Nn * D_LAT;
        init_bias<<<dim3((total + 255) / 256), blk, 0, stream>>>(out, b2, total, D_LAT - 1);
    }
    // 6) out[dst] += w * s2[src]
    {
        int d4shift = __builtin_ctz(D_LAT >> 2);           // 6
        int total = nE << d4shift;
        scatter_edges<<<dim3((total + 255) / 256), blk, 0, stream>>>(
            s_buf, src, dst, ew, out, nE, d4shift);
    }
    // 7) final ReLU in place
    {
        int total = Nn * D_LAT;
        relu_inplace<<<dim3((total + 255) / 256), blk, 0, stream>>>(out, total);
    }
}